// LQNGNN_84619445666401
// MI455X (gfx1250) — compile-verified
//
#include <hip/hip_runtime.h>

// ---------------- problem constants (match reference) ----------------
#define H   64
#define NE  200000
#define NT  100000
#define NP  50000
#define ECC 1600000
#define EET 200000
#define ETP 100000

typedef __attribute__((ext_vector_type(2))) float v2f;
typedef __attribute__((ext_vector_type(8))) float v8f;

__device__ __forceinline__ float leakyf(float v) { return v > 0.f ? v : 0.1f * v; }

// ---------------- elementwise helpers ----------------
__global__ void k_zero(float* __restrict__ p, long n) {
    long i = (long)blockIdx.x * blockDim.x + threadIdx.x;
    if (i < n) p[i] = 0.f;
}

__global__ void k_add(float* __restrict__ dst, const float* __restrict__ src, long n) {
    long i = (long)blockIdx.x * blockDim.x + threadIdx.x;
    if (i < n) dst[i] += src[i];
}

__global__ void k_leaky_copy(float* __restrict__ dst, const float* __restrict__ src, long n) {
    long i = (long)blockIdx.x * blockDim.x + threadIdx.x;
    if (i < n) dst[i] = leakyf(src[i]);
}

// ---------------- node encoders: y = leaky(x @ W^T + b), fin in {1,2,3} ----------------
// 8 threads per node, 8 output features per thread.
__global__ void k_encode(const float* __restrict__ x, const float* __restrict__ W,
                         const float* __restrict__ b, float* __restrict__ y,
                         int n, int fin) {
    long t = (long)blockIdx.x * blockDim.x + threadIdx.x;
    long node = t >> 3;
    int  o0   = ((int)t & 7) << 3;
    if (node >= n) return;
    float xi[4] = {0.f, 0.f, 0.f, 0.f};
    for (int f = 0; f < fin; ++f) xi[f] = x[node * fin + f];
    float* yo = y + node * H;
    #pragma unroll
    for (int o = 0; o < 8; ++o) {
        int oc = o0 + o;
        float s = b[oc];
        for (int f = 0; f < fin; ++f) s += xi[f] * W[oc * fin + f];
        yo[oc] = leakyf(s);
    }
}

// ---------------- fused edge encoder + scatter-add onto source entries ----------------
__global__ void k_edge_scatter(const float* __restrict__ ea, const float* __restrict__ W,
                               const float* __restrict__ b, const int* __restrict__ srcIdx,
                               float* __restrict__ agg) {
    long t = (long)blockIdx.x * blockDim.x + threadIdx.x;
    long e = t >> 3;
    int  o0 = ((int)t & 7) << 3;
    if (e >= ECC) return;
    int s = srcIdx[e];
    float x0 = ea[e * 4 + 0], x1 = ea[e * 4 + 1], x2 = ea[e * 4 + 2], x3 = ea[e * 4 + 3];
    float* dst = agg + (long)s * H;
    #pragma unroll
    for (int o = 0; o < 8; ++o) {
        int oc = o0 + o;
        float v = b[oc] + x0 * W[oc * 4 + 0] + x1 * W[oc * 4 + 1]
                        + x2 * W[oc * 4 + 2] + x3 * W[oc * 4 + 3];
        atomicAdd(dst + oc, leakyf(v));
    }
}

// ---------------- SAGE neighbor scatter: agg[dst] += x_src[src]; cnt[dst] += 1 ----------------
// 8 threads per edge, 8 floats each (two float4 loads).
__global__ void k_sage_scatter(const float* __restrict__ xsrc, const int* __restrict__ src,
                               const int* __restrict__ dst, int ne,
                               float* __restrict__ agg, float* __restrict__ cnt) {
    long t = (long)blockIdx.x * blockDim.x + threadIdx.x;
    long e = t >> 3;
    int  f0 = ((int)t & 7) << 3;
    if (e >= ne) return;
    __builtin_prefetch(src + e + 4096, 0, 0);   // global_prefetch_b8
    int s = src[e], d = dst[e];
    const float4* xs = (const float4*)(xsrc + (long)s * H + f0);
    float4 a = xs[0];
    float4 b = xs[1];
    float* ap = agg + (long)d * H + f0;
    atomicAdd(ap + 0, a.x); atomicAdd(ap + 1, a.y);
    atomicAdd(ap + 2, a.z); atomicAdd(ap + 3, a.w);
    atomicAdd(ap + 4, b.x); atomicAdd(ap + 5, b.y);
    atomicAdd(ap + 6, b.z); atomicAdd(ap + 7, b.w);
    if (f0 == 0) atomicAdd(cnt + d, 1.f);
}

// ---------------- WMMA SAGE linear: accum += (agg/cnt) @ Wl^T + bl + xdst @ Wr^T ----------------
// Fused as [agg/cnt | xdst] (nd x 128) @ B (128 x 64) via V_WMMA_F32_16X16X4_F32.
// Block = 128 threads = 4 waves; wave w owns rows [blk*64 + 16w, +16), full 64 cols.
// B is staged in LDS pre-paired: lB2[kp*LDSTR2 + n] = (B[2kp][n], B[2kp+1][n]), so each
// lane's B fragment is one aligned ds_load_b64 into an even VGPR pair (no mov packing).
// Row stride LDSTR2 = 80 float2 -> 2*80 mod 64 = 32: half-wave groups hit disjoint banks.
// Tail rows >= nd load from row 0 (address-safe); their products land only in D-rows the
// epilogue never stores, so no lane masking is needed in the K loop.
#define LDSTR2 80
__global__ void __launch_bounds__(128) k_sage_gemm(
    const float* __restrict__ agg, const float* __restrict__ cnt,
    const float* __restrict__ xdst, const float* __restrict__ Wl,
    const float* __restrict__ bl, const float* __restrict__ Wr,
    float* __restrict__ accum, int nd)
{
    __shared__ v2f   lB2[64 * LDSTR2];   // 40 KB
    __shared__ float lBias[64];
    int tid = threadIdx.x;
    {
        int k = tid;  // 0..127 : one K-row per thread
        const float* Wsel = (k < 64) ? (Wl + k) : (Wr + (k - 64));
        float* dstc = (float*)(lB2 + (k >> 1) * LDSTR2) + (k & 1);
        #pragma unroll 8
        for (int n = 0; n < 64; ++n) dstc[2 * n] = Wsel[n * 64];   // B[k][n]
        if (tid < 64) lBias[tid] = bl[tid];
    }
    __syncthreads();

    int wave = tid >> 5;
    int lane = tid & 31;
    int m    = lane & 15;            // A-matrix row within tile (also B column lane)
    int kb   = (lane >> 4) << 1;     // lanes 0-15 -> K 0,1 ; lanes 16-31 -> K 2,3
    long rowBase = (long)blockIdx.x * 64 + wave * 16;
    long row   = rowBase + m;
    long rsafe = row < nd ? row : 0;
    float cv  = cnt[rsafe];
    float inv = 1.f / (cv > 1.f ? cv : 1.f);
    const float* aRow = agg  + rsafe * H;
    const float* xRow = xdst + rsafe * H;

    v8f acc0 = {}, acc1 = {}, acc2 = {}, acc3 = {};

    // ---- K = 0..63 : normalized neighbor aggregate ----
    #pragma unroll 4
    for (int k0 = 0; k0 < 64; k0 += 4) {
        int ks = k0 + kb;                       // even -> 8B aligned
        v2f a = *(const v2f*)(aRow + ks);
        a = a * inv;
        const v2f* bp = lB2 + (ks >> 1) * LDSTR2 + m;
        v2f b0 = bp[0];
        v2f b1 = bp[16];
        v2f b2 = bp[32];
        v2f b3 = bp[48];
        acc0 = __builtin_amdgcn_wmma_f32_16x16x4_f32(false, a, false, b0, (short)0, acc0, false, false);
        acc1 = __builtin_amdgcn_wmma_f32_16x16x4_f32(false, a, false, b1, (short)0, acc1, false, false);
        acc2 = __builtin_amdgcn_wmma_f32_16x16x4_f32(false, a, false, b2, (short)0, acc2, false, false);
        acc3 = __builtin_amdgcn_wmma_f32_16x16x4_f32(false, a, false, b3, (short)0, acc3, false, false);
    }

    // ---- K = 64..127 : root (dst) features ----
    #pragma unroll 4
    for (int k0 = 0; k0 < 64; k0 += 4) {
        int ks = k0 + kb;
        v2f a = *(const v2f*)(xRow + ks);
        const v2f* bp = lB2 + ((64 + ks) >> 1) * LDSTR2 + m;
        v2f b0 = bp[0];
        v2f b1 = bp[16];
        v2f b2 = bp[32];
        v2f b3 = bp[48];
        acc0 = __builtin_amdgcn_wmma_f32_16x16x4_f32(false, a, false, b0, (short)0, acc0, false, false);
        acc1 = __builtin_amdgcn_wmma_f32_16x16x4_f32(false, a, false, b1, (short)0, acc1, false, false);
        acc2 = __builtin_amdgcn_wmma_f32_16x16x4_f32(false, a, false, b2, (short)0, acc2, false, false);
        acc3 = __builtin_amdgcn_wmma_f32_16x16x4_f32(false, a, false, b3, (short)0, acc3, false, false);
    }

    // C/D layout: VGPR v -> row rowBase + v (+8 for upper half-lanes), col = lane%16
    int cr   = (lane >> 4) * 8;
    int ccol = lane & 15;
    #pragma unroll
    for (int v = 0; v < 8; ++v) {
        long r = rowBase + cr + v;
        if (r < nd) {
            float* o = accum + r * H;
            o[ccol]      += acc0[v] + lBias[ccol];
            o[16 + ccol] += acc1[v] + lBias[16 + ccol];
            o[32 + ccol] += acc2[v] + lBias[32 + ccol];
            o[48 + ccol] += acc3[v] + lBias[48 + ccol];
        }
    }
}

// ---------------- prediction head: out = leaky(ex @ Wp^T + bp), Wp: [1,64] ----------------
__global__ void k_pred(const float* __restrict__ ex, const float* __restrict__ Wp,
                       const float* __restrict__ bp, float* __restrict__ out, int n) {
    int i = blockIdx.x * blockDim.x + threadIdx.x;
    if (i >= n) return;
    const float* r = ex + (long)i * H;
    float s = bp[0];
    #pragma unroll 16
    for (int k = 0; k < H; ++k) s += r[k] * Wp[k];
    out[i] = leakyf(s);
}

// ---------------- host side ----------------
static inline int blkc(long n, int t) { return (int)((n + t - 1) / t); }

extern "C" void kernel_launch(void* const* d_in, const int* in_sizes, int n_in,
                              void* d_out, int out_size, void* d_ws, size_t ws_size,
                              hipStream_t stream) {
    (void)in_sizes; (void)n_in; (void)out_size; (void)ws_size;

    // --- input mapping (setup_inputs() dict insertion order, tuples flattened) ---
    const float* entry_feats = (const float*)d_in[0];
    const float* task_feats  = (const float*)d_in[1];
    const float* proc_feats  = (const float*)d_in[2];
    const float* edge_attr   = (const float*)d_in[3];
    // params: enc_entry(W,b) enc_task(W,b) enc_proc(W,b) enc_edge(W,b) pred(W,b) -> 4..13
    // then 15 convs (Wl, bl, Wr) in CONV_NAMES order -> 14 + 3*j
    const float* P[64];
    for (int i = 4; i < 59 && i < n_in; ++i) P[i] = (const float*)d_in[i];
    const int* eiC = (const int*)d_in[59];  // [2, ECC]
    const int* eiT = (const int*)d_in[60];  // [2, EET]
    const int* eiP = (const int*)d_in[61];  // [2, ETP]
    const int *c0 = eiC, *c1 = eiC + ECC;
    const int *t0 = eiT, *t1 = eiT + EET;
    const int *p0 = eiP, *p1 = eiP + ETP;

    // --- workspace carve-up (floats) ---
    float* ws = (float*)d_ws;
    float* ex     = ws;                         // NE*H
    float* tx     = ex + (long)NE * H;          // NT*H
    float* px     = tx + (long)NT * H;          // NP*H
    float* agg_ee = px + (long)NP * H;          // NE*H
    float* agg    = agg_ee + (long)NE * H;      // NE*H (max of NE/NT/NP)
    float* cnt    = agg + (long)NE * H;         // NE
    float* accum  = cnt + NE;                   // NE*H

    const long NEH = (long)NE * H, NTH = (long)NT * H, NPH = (long)NP * H;

    // conv: SAGE(x_src -> agg over (src,dst)) ; accum += norm(agg)@Wl^T + bl + x_dst@Wr^T
    auto conv = [&](const float* xsrc, const float* xd, int nd,
                    const int* sIdx, const int* dIdx, int ne, int j) {
        const float* Wl = P[14 + 3 * j + 0];
        const float* bl = P[14 + 3 * j + 1];
        const float* Wr = P[14 + 3 * j + 2];
        long ndh = (long)nd * H;
        k_zero<<<blkc(ndh, 256), 256, 0, stream>>>(agg, ndh);
        k_zero<<<blkc(nd, 256), 256, 0, stream>>>(cnt, nd);
        k_sage_scatter<<<blkc((long)ne * 8, 256), 256, 0, stream>>>(xsrc, sIdx, dIdx, ne, agg, cnt);
        k_sage_gemm<<<blkc(nd, 64), 128, 0, stream>>>(agg, cnt, xd, Wl, bl, Wr, accum, nd);
    };

    // --- encoders ---
    k_encode<<<blkc((long)NE * 8, 256), 256, 0, stream>>>(entry_feats, P[4], P[5], ex, NE, 2);
    k_encode<<<blkc((long)NT * 8, 256), 256, 0, stream>>>(task_feats,  P[6], P[7], tx, NT, 3);
    k_encode<<<blkc((long)NP * 8, 256), 256, 0, stream>>>(proc_feats,  P[8], P[9], px, NP, 1);

    // --- agg_ee = scatter(leaky(enc_edge(edge_attr)), ei_calls[0]) ---
    k_zero<<<blkc(NEH, 256), 256, 0, stream>>>(agg_ee, NEH);
    k_edge_scatter<<<blkc((long)ECC * 8, 256), 256, 0, stream>>>(edge_attr, P[10], P[11], c0, agg_ee);

    // --- message-passing iterations ---
    for (int i = 0; i < 2; ++i) {
        int jb = 6 * i;  // conv param block for this iteration
        k_add<<<blkc(NEH, 256), 256, 0, stream>>>(ex, agg_ee, NEH);
        k_zero<<<blkc(NEH, 256), 256, 0, stream>>>(accum, NEH);
        conv(tx, ex, NE, t1, t0, EET, jb + 0);   // t2e
        conv(ex, ex, NE, c0, c1, ECC, jb + 1);   // e2e
        conv(ex, ex, NE, c1, c0, ECC, jb + 2);   // e2er
        k_leaky_copy<<<blkc(NEH, 256), 256, 0, stream>>>(ex, accum, NEH);

        k_zero<<<blkc(NTH, 256), 256, 0, stream>>>(accum, NTH);
        conv(ex, tx, NT, t0, t1, EET, jb + 3);   // e2t
        conv(px, tx, NT, p1, p0, ETP, jb + 4);   // p2t
        k_leaky_copy<<<blkc(NTH, 256), 256, 0, stream>>>(tx, accum, NTH);

        k_zero<<<blkc(NPH, 256), 256, 0, stream>>>(accum, NPH);
        conv(tx, px, NP, p0, p1, ETP, jb + 5);   // t2p
        k_leaky_copy<<<blkc(NPH, 256), 256, 0, stream>>>(px, accum, NPH);
    }

    // --- final block ---
    k_add<<<blkc(NEH, 256), 256, 0, stream>>>(ex, agg_ee, NEH);
    k_zero<<<blkc(NEH, 256), 256, 0, stream>>>(accum, NEH);
    conv(tx, ex, NE, t1, t0, EET, 12);           // fin_t2e
    conv(ex, ex, NE, c0, c1, ECC, 13);           // fin_e2e
    conv(ex, ex, NE, c1, c0, ECC, 14);           // fin_e2er
    k_leaky_copy<<<blkc(NEH, 256), 256, 0, stream>>>(ex, accum, NEH);

    // --- prediction head ---
    k_pred<<<blkc(NE, 256), 256, 0, stream>>>(ex, P[12], P[13], (float*)d_out, NE);
}